// WSkipGNN_33019708572410
// MI455X (gfx1250) — compile-verified
//
#include <hip/hip_runtime.h>

// WSkipGNN on MI455X (gfx1250, wave32).
// Scatter-bound workload; h (25.6MB) fits in 192MB L2 so edge traffic stays
// on-chip. Dense GEMMs use V_WMMA_F32_16X16X4_F32 (exact fp32 path).

typedef __attribute__((ext_vector_type(2))) float v2f;
typedef __attribute__((ext_vector_type(8))) float v8f;

#define HD 128

// ---------------- degree / normalization ----------------

__global__ __launch_bounds__(256)
void deg_init_kernel(float* __restrict__ deg, int n) {
  int t = blockIdx.x * 256 + threadIdx.x;
  if (t < n) deg[t] = 1.0f;  // self loop contribution
}

__global__ __launch_bounds__(256)
void deg_count_kernel(float* __restrict__ deg, const long long* __restrict__ dst, int e) {
  int t = blockIdx.x * 256 + threadIdx.x;
  if (t < e) atomicAdd(&deg[(int)dst[t]], 1.0f);
}

__global__ __launch_bounds__(256)
void deg_rsqrt_kernel(float* __restrict__ deg, int n) {
  int t = blockIdx.x * 256 + threadIdx.x;
  if (t < n) deg[t] = rsqrtf(deg[t]);  // deg >= 1 always
}

// ---------------- dense GEMM: Out[N,128] = A[N,128] @ W[128,128] ----------------
// 256 threads = 8 waves; block owns a 16-row strip, wave w owns cols [16w,16w+16).
// A tile staged in LDS with padded stride 132 (4-bank skew -> conflict free).
// Per k-step of 4: lanes 0-15 carry K={k0,k0+1}, lanes 16-31 carry K={k0+2,k0+3}
// for both A (16x4) and B (4x16) fragments per ISA 7.12.2 layouts.

__global__ __launch_bounds__(256)
void gemm_h128_kernel(const float* __restrict__ A, const float* __restrict__ W,
                      float* __restrict__ Out, int nrows) {
  __shared__ float As[16 * 132];
  int row0 = blockIdx.x * 16;
  for (int i = threadIdx.x; i < 16 * HD; i += 256) {
    int r = i >> 7, k = i & (HD - 1);
    int row = row0 + r;
    As[r * 132 + k] = (row < nrows) ? A[(size_t)row * HD + k] : 0.0f;
  }
  __syncthreads();

  int lane = threadIdx.x & 31;
  int wave = threadIdx.x >> 5;
  int half = lane >> 4;       // 0: K pair {k0,k0+1}, 1: {k0+2,k0+3}
  int mn   = lane & 15;       // row (A) / col (B) index within tile
  int col  = wave * 16 + mn;

  v8f acc = {};
#pragma unroll
  for (int k0 = 0; k0 < HD; k0 += 4) {
    int ka = k0 + 2 * half;
    v2f a, b;
    a.x = As[mn * 132 + ka];
    a.y = As[mn * 132 + ka + 1];
    b.x = W[(size_t)ka * HD + col];        // W fits in cache (64KB)
    b.y = W[(size_t)(ka + 1) * HD + col];
    acc = __builtin_amdgcn_wmma_f32_16x16x4_f32(false, a, false, b,
                                                (short)0, acc, false, false);
  }
  // D layout: VGPR r -> row r (lanes 0-15) / row 8+r (lanes 16-31), col = mn
#pragma unroll
  for (int r = 0; r < 8; ++r) {
    int row = row0 + r + 8 * half;
    if (row < nrows) Out[(size_t)row * HD + col] = acc[r];
  }
}

// ---------------- aggregation ----------------

// agg[n,:] = bias + h[n,:] * dinv[n]^2   (self loop folded in)
__global__ __launch_bounds__(256)
void agg_init_kernel(float* __restrict__ agg, const float* __restrict__ h,
                     const float* __restrict__ dinv, const float* __restrict__ bias,
                     int n32) {
  int t = blockIdx.x * 256 + threadIdx.x;
  if (t >= n32) return;
  int node = t >> 5, c = (t & 31) << 2;
  float di = dinv[node];
  float s = di * di;
  float4 hv = *(const float4*)(h + (size_t)node * HD + c);
  float4 o;
  o.x = bias[c + 0] + hv.x * s;
  o.y = bias[c + 1] + hv.y * s;
  o.z = bias[c + 2] + hv.z * s;
  o.w = bias[c + 3] + hv.w * s;
  *(float4*)(agg + (size_t)node * HD + c) = o;
}

// agg[dst,:] += h[src,:] * dinv[src]*dinv[dst]   (32 lanes per edge, float4 each)
__global__ __launch_bounds__(256)
void agg_edges_kernel(float* __restrict__ agg, const float* __restrict__ h,
                      const float* __restrict__ dinv,
                      const long long* __restrict__ src,
                      const long long* __restrict__ dst, int e32) {
  int t = blockIdx.x * 256 + threadIdx.x;
  if (t >= e32) return;
  int eid = t >> 5, c = (t & 31) << 2;
  int s = (int)src[eid];
  int d = (int)dst[eid];
  float w = dinv[s] * dinv[d];
  float4 hv = *(const float4*)(h + (size_t)s * HD + c);
  float* ap = agg + (size_t)d * HD + c;
  atomicAdd(ap + 0, hv.x * w);
  atomicAdd(ap + 1, hv.y * w);
  atomicAdd(ap + 2, hv.z * w);
  atomicAdd(ap + 3, hv.w * w);
}

// in place: agg = relu(agg) [+ res_weight * xprev]
__global__ __launch_bounds__(256)
void finish_kernel(float* __restrict__ agg, const float* __restrict__ xprev,
                   const float* __restrict__ rw_ptr, int use_res, int n4) {
  int t = blockIdx.x * 256 + threadIdx.x;
  if (t >= n4) return;
  float4 v = ((const float4*)agg)[t];
  v.x = fmaxf(v.x, 0.0f);
  v.y = fmaxf(v.y, 0.0f);
  v.z = fmaxf(v.z, 0.0f);
  v.w = fmaxf(v.w, 0.0f);
  if (use_res) {
    float rw = rw_ptr[0];
    float4 p = ((const float4*)xprev)[t];
    v.x += rw * p.x;
    v.y += rw * p.y;
    v.z += rw * p.z;
    v.w += rw * p.w;
  }
  ((float4*)agg)[t] = v;
}

// ---------------- MLP head: Out[N,40] = X[N,128] @ W[128,40] + b ----------------
// 96 threads = 3 waves; cols padded to 48 in LDS (zero fill), masked store.

__global__ __launch_bounds__(96)
void mlp_kernel(const float* __restrict__ X, const float* __restrict__ W,
                const float* __restrict__ b, float* __restrict__ Out,
                int nrows, int ncols) {
  __shared__ float As[16 * 132];
  __shared__ float Wt[48 * 132];  // Wt[n][k] = W[k][n], zero for n >= ncols
  int row0 = blockIdx.x * 16;
  for (int i = threadIdx.x; i < 48 * HD; i += 96) {
    int n = i >> 7, k = i & (HD - 1);
    Wt[n * 132 + k] = (n < ncols) ? W[(size_t)k * ncols + n] : 0.0f;
  }
  for (int i = threadIdx.x; i < 16 * HD; i += 96) {
    int r = i >> 7, k = i & (HD - 1);
    int row = row0 + r;
    As[r * 132 + k] = (row < nrows) ? X[(size_t)row * HD + k] : 0.0f;
  }
  __syncthreads();

  int lane = threadIdx.x & 31;
  int wave = threadIdx.x >> 5;
  int half = lane >> 4;
  int mn   = lane & 15;
  int col  = wave * 16 + mn;

  v8f acc = {};
#pragma unroll
  for (int k0 = 0; k0 < HD; k0 += 4) {
    int ka = k0 + 2 * half;
    v2f a, bb;
    a.x  = As[mn * 132 + ka];
    a.y  = As[mn * 132 + ka + 1];
    bb.x = Wt[col * 132 + ka];
    bb.y = Wt[col * 132 + ka + 1];
    acc = __builtin_amdgcn_wmma_f32_16x16x4_f32(false, a, false, bb,
                                                (short)0, acc, false, false);
  }
  if (col < ncols) {
    float bv = b[col];
#pragma unroll
    for (int r = 0; r < 8; ++r) {
      int row = row0 + r + 8 * half;
      if (row < nrows) Out[(size_t)row * ncols + col] = acc[r] + bv;
    }
  }
}

// ---------------- host launcher ----------------

extern "C" void kernel_launch(void* const* d_in, const int* in_sizes, int n_in,
                              void* d_out, int out_size, void* d_ws, size_t ws_size,
                              hipStream_t stream) {
  (void)n_in; (void)out_size; (void)ws_size;
  const float*     X    = (const float*)d_in[0];
  const long long* ei   = (const long long*)d_in[1];  // int64 [2, E]
  const float*     Ws   = (const float*)d_in[2];      // [L,128,128]
  const float*     bs   = (const float*)d_in[3];      // [L,128]
  const float*     mlpW = (const float*)d_in[4];      // [128,C]
  const float*     mlpb = (const float*)d_in[5];      // [C]
  const float*     rw   = (const float*)d_in[6];      // scalar
  float*           out  = (float*)d_out;

  const int N = in_sizes[0] / HD;   // 50000
  const int E = in_sizes[1] / 2;    // 800000
  const int C = in_sizes[5];        // 40
  const int L = in_sizes[3] / HD;   // 4

  // workspace layout: dinv[N] | bufA[N*128] | bufB[N*128] | bufH[N*128]
  char* ws = (char*)d_ws;
  size_t off = ((size_t)N * sizeof(float) + 255) & ~(size_t)255;
  const size_t nh = (size_t)N * HD * sizeof(float);
  float* dinv = (float*)ws;
  float* bufA = (float*)(ws + off);
  float* bufB = (float*)(ws + off + nh);
  float* bufH = (float*)(ws + off + 2 * nh);

  const long long* srcI = ei;
  const long long* dstI = ei + E;

  dim3 blk(256);
  deg_init_kernel<<<(N + 255) / 256, blk, 0, stream>>>(dinv, N);
  deg_count_kernel<<<(E + 255) / 256, blk, 0, stream>>>(dinv, dstI, E);
  deg_rsqrt_kernel<<<(N + 255) / 256, blk, 0, stream>>>(dinv, N);

  const int gemmGrid = (N + 15) / 16;
  const int n32 = N * 32;            // N*128/4 float4 units
  const int e32 = E * 32;
  const int g_n32 = (n32 + 255) / 256;
  const int g_e32 = (e32 + 255) / 256;

  const float* xcur = X;
  float* pingpong[2] = {bufA, bufB};
  for (int l = 0; l < L; ++l) {
    float* agg = pingpong[l & 1];
    gemm_h128_kernel<<<gemmGrid, blk, 0, stream>>>(xcur, Ws + (size_t)l * HD * HD,
                                                   bufH, N);
    agg_init_kernel<<<g_n32, blk, 0, stream>>>(agg, bufH, dinv,
                                               bs + (size_t)l * HD, n32);
    agg_edges_kernel<<<g_e32, blk, 0, stream>>>(agg, bufH, dinv, srcI, dstI, e32);
    if (l < L - 1) {  // relu (+ residual for middle layers); last conv is linear
      finish_kernel<<<g_n32, blk, 0, stream>>>(agg, xcur, rw, (l > 0) ? 1 : 0, n32);
    }
    xcur = agg;
  }

  mlp_kernel<<<gemmGrid, dim3(96), 0, stream>>>(xcur, mlpW, mlpb, out, N, C);
}